// HGT_72232759984501
// MI455X (gfx1250) — compile-verified
//
#include <hip/hip_runtime.h>
#include <math.h>

// ---------------- problem constants ----------------
static constexpr int  kNA   = 100000;
static constexpr int  kNP   = 200000;
static constexpr int  kFEAT = 128;
static constexpr int  kD    = 32;
static constexpr int  kEW   = 1000000;
static constexpr int  kEWB  = 1000000;
static constexpr int  kEC   = 2000000;

typedef __attribute__((ext_vector_type(2))) float v2f;
typedef __attribute__((ext_vector_type(8))) float v8f;

__device__ __forceinline__ float gelu_exact(float x) {
  return 0.5f * x * (1.0f + erff(x * 0.70710678118654752440f));
}

// float atomic-max via monotone integer encoding (init memory to 0xFF800000 = -inf)
__device__ __forceinline__ void atomic_max_float(unsigned int* addr, float v) {
  if (v >= 0.0f) atomicMax((int*)addr, __float_as_int(v));
  else           atomicMin(addr, __float_as_uint(v));
}

// ---------------- generic fill ----------------
__global__ void fill_u32_kernel(unsigned int* __restrict__ p, unsigned int val, long n) {
  long i = (long)blockIdx.x * blockDim.x + threadIdx.x;
  long stride = (long)gridDim.x * blockDim.x;
  for (; i < n; i += stride) p[i] = val;
}

// ---------------- fused k/q/v projection: 3 GEMMs sharing A-fragment loads ----------------
// kout/qout/vout[N x 32] = X[N x din] @ {Wk,Wq,Wv}[din x 32] + {bk,bq,bv}
__global__ void kqv_wmma_kernel(const float* __restrict__ X,
                                const float* __restrict__ Wk, const float* __restrict__ bk,
                                const float* __restrict__ Wq, const float* __restrict__ bq,
                                const float* __restrict__ Wv, const float* __restrict__ bv,
                                float* __restrict__ kout, float* __restrict__ qout,
                                float* __restrict__ vout, int N, int din) {
  const int lane = threadIdx.x & 31;
  const int wave = threadIdx.x >> 5;
  const int tile = blockIdx.x * (blockDim.x >> 5) + wave;
  const int row0 = tile * 16;
  if (row0 >= N) return;                 // whole-wave exit: EXEC all-ones at WMMA
  const int m  = lane & 15;
  const int hi = lane >> 4;
  int rowA = row0 + m; if (rowA >= N) rowA = N - 1;
  const float* xr = X + (long)rowA * din;

  v8f ak0 = {0,0,0,0,0,0,0,0}, ak1 = ak0;
  v8f aq0 = ak0, aq1 = ak0, av0 = ak0, av1 = ak0;

  for (int k0 = 0; k0 < din; k0 += 4) {
    const int kb = k0 + 2 * hi;
    v2f a; a[0] = xr[kb]; a[1] = xr[kb + 1];
    const long wo = (long)kb * 32;
    v2f b0, b1;
    b0[0] = Wk[wo + m];      b0[1] = Wk[wo + 32 + m];
    b1[0] = Wk[wo + 16 + m]; b1[1] = Wk[wo + 48 + m];
    ak0 = __builtin_amdgcn_wmma_f32_16x16x4_f32(false, a, false, b0, (short)0, ak0, false, false);
    ak1 = __builtin_amdgcn_wmma_f32_16x16x4_f32(false, a, false, b1, (short)0, ak1, false, false);
    b0[0] = Wq[wo + m];      b0[1] = Wq[wo + 32 + m];
    b1[0] = Wq[wo + 16 + m]; b1[1] = Wq[wo + 48 + m];
    aq0 = __builtin_amdgcn_wmma_f32_16x16x4_f32(false, a, false, b0, (short)0, aq0, false, false);
    aq1 = __builtin_amdgcn_wmma_f32_16x16x4_f32(false, a, false, b1, (short)0, aq1, false, false);
    b0[0] = Wv[wo + m];      b0[1] = Wv[wo + 32 + m];
    b1[0] = Wv[wo + 16 + m]; b1[1] = Wv[wo + 48 + m];
    av0 = __builtin_amdgcn_wmma_f32_16x16x4_f32(false, a, false, b0, (short)0, av0, false, false);
    av1 = __builtin_amdgcn_wmma_f32_16x16x4_f32(false, a, false, b1, (short)0, av1, false, false);
  }

  const float bk0 = bk[m], bk1 = bk[16 + m];
  const float bq0 = bq[m], bq1 = bq[16 + m];
  const float bv0 = bv[m], bv1 = bv[16 + m];
  #pragma unroll
  for (int g = 0; g < 8; ++g) {
    int row = row0 + g + 8 * hi;
    if (row < N) {
      long base = (long)row * 32;
      kout[base + m]      = ak0[g] + bk0;
      kout[base + 16 + m] = ak1[g] + bk1;
      qout[base + m]      = aq0[g] + bq0;
      qout[base + 16 + m] = aq1[g] + bq1;
      vout[base + m]      = av0[g] + bv0;
      vout[base + 16 + m] = av1[g] + bv1;
    }
  }
}

// ---------------- generic WMMA GEMM (templated activation / skip epilogue) ----------------
// C[N x 32] = act(X[N x din]) @ W[din x 32] (+bias) (then optional gated skip)
template <bool ACT, bool SKIP>
__global__ void gemm32_wmma_kernel(const float* __restrict__ X,
                                   const float* __restrict__ W,
                                   const float* __restrict__ bias,
                                   float* __restrict__ C,
                                   int N, int din,
                                   const float* __restrict__ skipX,
                                   const float* __restrict__ skipGate) {
  const int lane = threadIdx.x & 31;
  const int wave = threadIdx.x >> 5;
  const int tile = blockIdx.x * (blockDim.x >> 5) + wave;
  const int row0 = tile * 16;
  if (row0 >= N) return;
  const int m  = lane & 15;
  const int hi = lane >> 4;
  int rowA = row0 + m; if (rowA >= N) rowA = N - 1;
  const float* xr = X + (long)rowA * din;

  v8f acc0 = {0,0,0,0,0,0,0,0};
  v8f acc1 = acc0;

  for (int k0 = 0; k0 < din; k0 += 8) {   // unroll K by 8: two 16x16x4 steps per trip
    #pragma unroll
    for (int u = 0; u < 2; ++u) {
      const int kb = k0 + 4 * u + 2 * hi;
      float a0 = xr[kb], a1 = xr[kb + 1];
      if (ACT) { a0 = gelu_exact(a0); a1 = gelu_exact(a1); }
      v2f a; a[0] = a0; a[1] = a1;
      const float* w0 = W + (long)kb * 32;
      v2f b0, b1;
      b0[0] = w0[m];       b0[1] = w0[32 + m];
      b1[0] = w0[16 + m];  b1[1] = w0[48 + m];
      acc0 = __builtin_amdgcn_wmma_f32_16x16x4_f32(false, a, false, b0, (short)0, acc0, false, false);
      acc1 = __builtin_amdgcn_wmma_f32_16x16x4_f32(false, a, false, b1, (short)0, acc1, false, false);
    }
  }

  float gate = 1.0f, gcomp = 0.0f;
  if (SKIP) { gate = 1.0f / (1.0f + __expf(-skipGate[0])); gcomp = 1.0f - gate; }
  const float bia0 = bias ? bias[m]      : 0.0f;
  const float bia1 = bias ? bias[16 + m] : 0.0f;

  #pragma unroll
  for (int g = 0; g < 8; ++g) {
    int row = row0 + g + 8 * hi;
    if (row < N) {
      long base = (long)row * 32;
      float o0 = acc0[g] + bia0;
      float o1 = acc1[g] + bia1;
      if (SKIP) {
        o0 = gate * o0 + gcomp * skipX[base + m];
        o1 = gate * o1 + gcomp * skipX[base + 16 + m];
      }
      C[base + m]      = o0;
      C[base + 16 + m] = o1;
    }
  }
}

// ---------------- edge phase (wave per edge; coalesced 128B row gathers) ----------------
__global__ void edge_score_max_kernel(const int* __restrict__ edges, int E,
                                      const float* __restrict__ q,
                                      const float* __restrict__ krel,
                                      const float* __restrict__ prel_r, float scale,
                                      float* __restrict__ score,
                                      unsigned int* __restrict__ smax) {
  const int lane = threadIdx.x & 31;
  const int e = blockIdx.x * (blockDim.x >> 5) + (threadIdx.x >> 5);
  if (e >= E) return;
  const int sIdx = edges[e];
  const int dIdx = edges[E + e];
  float p = q[(long)dIdx * 32 + lane] * krel[(long)sIdx * 32 + lane];
  for (int off = 16; off > 0; off >>= 1) p += __shfl_xor(p, off, 32);
  const float sc = p * prel_r[0] * scale;
  if (lane == 0) {
    score[e] = sc;
    atomic_max_float(smax + dIdx, sc);
  }
}

__global__ void edge_exp_den_kernel(const int* __restrict__ edges, int E,
                                    const float* __restrict__ score,
                                    const unsigned int* __restrict__ smax,
                                    float* __restrict__ den) {
  const int e = blockIdx.x * blockDim.x + threadIdx.x;
  if (e >= E) return;
  const int dIdx = edges[E + e];
  const float ex = __expf(score[e] - __uint_as_float(smax[dIdx]));
  atomicAdd(den + dIdx, ex);
}

__global__ void edge_scatter_kernel(const int* __restrict__ edges, int E,
                                    const float* __restrict__ score,
                                    const unsigned int* __restrict__ smax,
                                    const float* __restrict__ den,
                                    const float* __restrict__ vrel,
                                    float* __restrict__ out) {
  const int lane = threadIdx.x & 31;
  const int e = blockIdx.x * (blockDim.x >> 5) + (threadIdx.x >> 5);
  if (e >= E) return;
  const int sIdx = edges[e];
  const int dIdx = edges[E + e];
  const float alpha = __expf(score[e] - __uint_as_float(smax[dIdx])) / den[dIdx];
  atomicAdd(out + (long)dIdx * 32 + lane, vrel[(long)sIdx * 32 + lane] * alpha);
}

// ---------------- host-side orchestration ----------------
static inline int gemm_blocks(int N) { return ((N + 15) / 16 + 7) / 8; }

static inline void launch_gemm(hipStream_t s, const float* X, const float* W, const float* bias,
                               float* C, int N, int din, bool act,
                               const float* skipX, const float* skipGate) {
  const int blocks = gemm_blocks(N);
  if (act && skipX)
    gemm32_wmma_kernel<true, true><<<blocks, 256, 0, s>>>(X, W, bias, C, N, din, skipX, skipGate);
  else if (act)
    gemm32_wmma_kernel<true, false><<<blocks, 256, 0, s>>>(X, W, bias, C, N, din, nullptr, nullptr);
  else
    gemm32_wmma_kernel<false, false><<<blocks, 256, 0, s>>>(X, W, bias, C, N, din, nullptr, nullptr);
}

static inline void launch_fill(hipStream_t s, void* p, unsigned int v, long n) {
  long b = (n + 255) / 256; if (b > 65535) b = 65535;
  fill_u32_kernel<<<(unsigned)b, 256, 0, s>>>((unsigned int*)p, v, n);
}

struct RelDesc { const int* e; int E; int st; int dt; };

static void run_layer(hipStream_t s,
                      const float* xA, const float* xP, int din,
                      const int* eW, const int* eWB, const int* eCi,
                      const float* Wk, const float* bk,
                      const float* Wq, const float* bq,
                      const float* Wv, const float* bv,
                      const float* Wa, const float* ba,
                      const float* skip, const float* arel, const float* mrel,
                      const float* prel, bool use_skip,
                      float* kA, float* qA, float* vA,
                      float* kP, float* qP, float* vP,
                      float* outA, float* outP,
                      float* krel, float* vrel, float* score,
                      unsigned int* smax, float* den,
                      float* resA, float* resP) {
  const long WT = (long)din * 32;   // per-type stride inside Wk/Wq/Wv
  // fused per-type k/q/v projections (6 WMMAs per shared A fragment)
  kqv_wmma_kernel<<<gemm_blocks(kNA), 256, 0, s>>>(xA, Wk, bk, Wq, bq, Wv, bv,
                                                   kA, qA, vA, kNA, din);
  kqv_wmma_kernel<<<gemm_blocks(kNP), 256, 0, s>>>(xP, Wk + WT, bk + 32, Wq + WT, bq + 32,
                                                   Wv + WT, bv + 32, kP, qP, vP, kNP, din);
  // zero aggregation buffers
  launch_fill(s, outA, 0u, (long)kNA * 32);
  launch_fill(s, outP, 0u, (long)kNP * 32);

  const RelDesc rels[3] = { {eW, kEW, 0, 1}, {eWB, kEWB, 1, 0}, {eCi, kEC, 1, 1} };
  const float scale = 0.17677669529663687f;   // 1/sqrt(32)
  for (int r = 0; r < 3; ++r) {
    const float* ksrc = rels[r].st ? kP : kA;
    const float* vsrc = rels[r].st ? vP : vA;
    const int    Ns   = rels[r].st ? kNP : kNA;
    const float* qdst = rels[r].dt ? qP : qA;
    const int    Nd   = rels[r].dt ? kNP : kNA;
    float*       outd = rels[r].dt ? outP : outA;
    // relation-specific transforms (WMMA, 32x32)
    launch_gemm(s, ksrc, arel + (long)r * 1024, nullptr, krel, Ns, 32, false, nullptr, nullptr);
    launch_gemm(s, vsrc, mrel + (long)r * 1024, nullptr, vrel, Ns, 32, false, nullptr, nullptr);
    launch_fill(s, smax, 0xFF800000u, Nd);    // -inf
    launch_fill(s, den,  0u,          Nd);
    const int E = rels[r].E;
    const int wblocks = (E + 7) / 8;
    edge_score_max_kernel<<<wblocks, 256, 0, s>>>(rels[r].e, E, qdst, krel, prel + r, scale, score, smax);
    edge_exp_den_kernel<<<(E + 255) / 256, 256, 0, s>>>(rels[r].e, E, score, smax, den);
    edge_scatter_kernel<<<wblocks, 256, 0, s>>>(rels[r].e, E, score, smax, den, vrel, outd);
  }
  // gelu + output linear (+ gated skip on layer 2)
  launch_gemm(s, outA, Wa,        ba,      resA, kNA, 32, true,
              use_skip ? xA : nullptr, use_skip ? skip : nullptr);
  launch_gemm(s, outP, Wa + 1024, ba + 32, resP, kNP, 32, true,
              use_skip ? xP : nullptr, use_skip ? skip + 1 : nullptr);
}

extern "C" void kernel_launch(void* const* d_in, const int* in_sizes, int n_in,
                              void* d_out, int out_size, void* d_ws, size_t ws_size,
                              hipStream_t stream) {
  (void)in_sizes; (void)n_in; (void)out_size; (void)ws_size;
  const float* xA    = (const float*)d_in[0];
  const float* xP    = (const float*)d_in[1];
  const int*   eW    = (const int*)d_in[2];
  const int*   eWB   = (const int*)d_in[3];
  const int*   eCi   = (const int*)d_in[4];
  const float* Wk1   = (const float*)d_in[5],  *bk1 = (const float*)d_in[6];
  const float* Wq1   = (const float*)d_in[7],  *bq1 = (const float*)d_in[8];
  const float* Wv1   = (const float*)d_in[9],  *bv1 = (const float*)d_in[10];
  const float* Wa1   = (const float*)d_in[11], *ba1 = (const float*)d_in[12];
  const float* skip1 = (const float*)d_in[13];
  const float* arel1 = (const float*)d_in[14], *mrel1 = (const float*)d_in[15], *prel1 = (const float*)d_in[16];
  const float* Wk2   = (const float*)d_in[17], *bk2 = (const float*)d_in[18];
  const float* Wq2   = (const float*)d_in[19], *bq2 = (const float*)d_in[20];
  const float* Wv2   = (const float*)d_in[21], *bv2 = (const float*)d_in[22];
  const float* Wa2   = (const float*)d_in[23], *ba2 = (const float*)d_in[24];
  const float* skip2 = (const float*)d_in[25];
  const float* arel2 = (const float*)d_in[26], *mrel2 = (const float*)d_in[27], *prel2 = (const float*)d_in[28];

  // workspace carve-up (floats)
  float* ws = (float*)d_ws;
  const long NA32 = (long)kNA * 32, NP32 = (long)kNP * 32;
  float* kA   = ws;            float* qA = kA + NA32;   float* vA = qA + NA32;
  float* kP   = vA + NA32;     float* qP = kP + NP32;   float* vP = qP + NP32;
  float* x1A  = vP + NP32;     float* x1P = x1A + NA32;          // layer-1 residuals
  float* outA = x1P + NP32;    float* outP = outA + NA32;        // aggregation buffers
  float* krel = outP + NP32;   float* vrel = krel + NP32;        // per-relation (max = papers)
  float* score = vrel + NP32;                                    // per-edge scores (max 2M)
  unsigned int* smax = (unsigned int*)(score + kEC);
  float* den = (float*)(smax + kNP);

  float* outAuth = (float*)d_out;
  float* outPap  = outAuth + NA32;

  // layer 1: 128 -> 32, no skip
  run_layer(stream, xA, xP, kFEAT, eW, eWB, eCi,
            Wk1, bk1, Wq1, bq1, Wv1, bv1, Wa1, ba1, skip1, arel1, mrel1, prel1, false,
            kA, qA, vA, kP, qP, vP, outA, outP, krel, vrel, score, smax, den, x1A, x1P);
  // layer 2: 32 -> 32, gated skip; writes final outputs
  run_layer(stream, x1A, x1P, kD, eW, eWB, eCi,
            Wk2, bk2, Wq2, bq2, Wv2, bv2, Wa2, ba2, skip2, arel2, mrel2, prel2, true,
            kA, qA, vA, kP, qP, vP, outA, outP, krel, vrel, score, smax, den, outAuth, outPap);
}